// Attention_56676388438599
// MI455X (gfx1250) — compile-verified
//
#include <hip/hip_runtime.h>
#include <hip/hip_bf16.h>
#include <math.h>
#include <cstdint>

// ---------------------------------------------------------------------------
// Shapes: B=16, S=16, D=4096, H=32, KV=8, HD=128, N_REP=4, MAX_S=4096, START=4080
// ---------------------------------------------------------------------------

typedef float v2f __attribute__((ext_vector_type(2)));
typedef float v8f __attribute__((ext_vector_type(8)));

__device__ __forceinline__ v8f wmma4(v2f a, v2f b, v8f c) {
  return __builtin_amdgcn_wmma_f32_16x16x4_f32(
      false, a, false, b, (short)0, c, false, false);
}

// --- CDNA5 async copies: global -> LDS, tracked by ASYNCcnt ---
__device__ __forceinline__ void async_ld_b128(uint32_t ldsoff, const void* g) {
  asm volatile("global_load_async_to_lds_b128 %0, %1, off"
               :: "v"(ldsoff), "v"(g) : "memory");
}
__device__ __forceinline__ void async_ld_b128_nt(uint32_t ldsoff, const void* g) {
  asm volatile("global_load_async_to_lds_b128 %0, %1, off th:TH_LOAD_NT"
               :: "v"(ldsoff), "v"(g) : "memory");
}
__device__ __forceinline__ void async_ld_b32_nt(uint32_t ldsoff, const void* g) {
  asm volatile("global_load_async_to_lds_b32 %0, %1, off th:TH_LOAD_NT"
               :: "v"(ldsoff), "v"(g) : "memory");
}
__device__ __forceinline__ void wait_async0() {
  asm volatile("s_wait_asynccnt 0" ::: "memory");
}
__device__ __forceinline__ void wait_async8() {
  asm volatile("s_wait_asynccnt 8" ::: "memory");
}
__device__ __forceinline__ uint32_t lds_off(const void* p) {
  return (uint32_t)(uintptr_t)p;  // low 32 bits of generic LDS ptr = LDS offset
}

// ---------------------------------------------------------------------------
// GEMM partials: Yp[split][256,N] = X[256, krange] @ W[krange,N]
// grid = (N/64, 2). W staged TRANSPOSED in LDS so both A and B fragments are
// single contiguous b64 loads.
// ---------------------------------------------------------------------------
__global__ __launch_bounds__(256)
void gemm_wmma(const float* __restrict__ X, const float* __restrict__ W,
               float* __restrict__ Yp, int N) {
  constexpr int K = 4096, KC = 32, XS = KC + 4, WT = KC + 2, KHALF = 2048;
  __shared__ float x_lds[256 * XS];  // 36.9 KB, [row][k]
  __shared__ float w_lds[64 * WT];   //  8.7 KB, [col][k] (transposed)

  const int tid = threadIdx.x, lane = tid & 31;
  const int lmod = lane & 15, lhalf = lane >> 4, wave = tid >> 5;
  const int n0 = blockIdx.x * 64;
  const int kb = blockIdx.y * KHALF;
  const int m0a = wave * 32;
  float* Y = Yp + (size_t)blockIdx.y * 256 * N;

  v8f acc[2][4];
#pragma unroll
  for (int i = 0; i < 2; ++i)
#pragma unroll
    for (int j = 0; j < 4; ++j) acc[i][j] = v8f{};

  for (int k0 = kb; k0 < kb + KHALF; k0 += KC) {
    __syncthreads();
    // X chunk [256 x 32] -> LDS (8 async b128 per thread)
#pragma unroll
    for (int i = 0; i < 8; ++i) {
      int idx = tid + 256 * i;
      int row = idx >> 3, c4 = idx & 7;
      async_ld_b128(lds_off(&x_lds[row * XS + c4 * 4]),
                    X + (size_t)row * K + k0 + c4 * 4);
    }
    // W chunk [32 x 64] -> transposed LDS (8 async b32 per thread; coalesced
    // on the global side: consecutive lanes read consecutive columns)
#pragma unroll
    for (int i = 0; i < 8; ++i) {
      int e = tid + 256 * i;  // 0..2047
      int row = e >> 6, col = e & 63;
      async_ld_b32_nt(lds_off(&w_lds[col * WT + row]),
                      W + (size_t)(k0 + row) * N + n0 + col);
    }
    wait_async0();
    __syncthreads();

#pragma unroll
    for (int k4 = 0; k4 < KC; k4 += 4) {
      const int kk = k4 + 2 * lhalf;
      v2f a0 = *(const v2f*)&x_lds[(m0a + lmod) * XS + kk];
      v2f a1 = *(const v2f*)&x_lds[(m0a + 16 + lmod) * XS + kk];
#pragma unroll
      for (int nt = 0; nt < 4; ++nt) {
        v2f b = *(const v2f*)&w_lds[(nt * 16 + lmod) * WT + kk];
        acc[0][nt] = wmma4(a0, b, acc[0][nt]);
        acc[1][nt] = wmma4(a1, b, acc[1][nt]);
      }
    }
  }
#pragma unroll
  for (int nt = 0; nt < 4; ++nt)
#pragma unroll
    for (int r = 0; r < 8; ++r) {
      int row0 = m0a + r + lhalf * 8;
      Y[(size_t)row0 * N + n0 + nt * 16 + lmod] = acc[0][nt][r];
      Y[(size_t)(row0 + 16) * N + n0 + nt * 16 + lmod] = acc[1][nt][r];
    }
}

// ---------------------------------------------------------------------------
// K-split reductions (deterministic fixed-order adds)
// ---------------------------------------------------------------------------
__global__ void reduce_bias(const float* __restrict__ p,
                            const float* __restrict__ bias,
                            float* __restrict__ y, int N, int total) {
  int i = blockIdx.x * blockDim.x + threadIdx.x;
  if (i >= total) return;
  y[i] = p[i] + p[total + i] + bias[i & (N - 1)];
}

__global__ void reduce_rope(const float* __restrict__ p,
                            const float* __restrict__ bias,
                            float* __restrict__ y, const float* __restrict__ fc,
                            const float* __restrict__ fs, int nheads) {
  int idx = blockIdx.x * blockDim.x + threadIdx.x;
  int total_pairs = 256 * nheads * 64;
  if (idx >= total_pairs) return;
  int j = idx & 63;
  int h = (idx >> 6) % nheads;
  int s = (idx / (64 * nheads)) & 15;
  int total = 256 * nheads * 128;
  int e0 = idx * 2;
  float xr = p[e0] + p[total + e0] + bias[h * 128 + 2 * j];
  float xi = p[e0 + 1] + p[total + e0 + 1] + bias[h * 128 + 2 * j + 1];
  float c = fc[s * 64 + j], sn = fs[s * 64 + j];
  y[e0] = xr * c - xi * sn;
  y[e0 + 1] = xr * sn + xi * c;
}

// ---------------------------------------------------------------------------
// Flash attention partials. grid = (B*KV=128, KSPLIT=2); 64 q rows per block.
// ---------------------------------------------------------------------------
__global__ __launch_bounds__(256)
void attn_kernel(const float* __restrict__ q, const float* __restrict__ kws,
                 const float* __restrict__ vws,
                 const float* __restrict__ cache_k,
                 const float* __restrict__ cache_v,
                 float* __restrict__ o_part, float* __restrict__ m_part,
                 float* __restrict__ l_part) {
  constexpr int QS = 132;  // Q/K LDS row stride
  constexpr int VS = 144;  // V LDS row stride
  constexpr int SS = 68;   // S/P LDS row stride
  extern __shared__ float smem[];
  float* q_lds = smem;             // 64 x 132
  float* k_lds = q_lds + 64 * QS;  // 64 x 132
  float* v_lds = k_lds + 64 * QS;  // 64 x 144
  float* s_lds = v_lds + 64 * VS;  // 64 x 68
  float* m_l = s_lds + 64 * SS;
  float* l_l = m_l + 64;
  float* c_l = l_l + 64;

  const int tid = threadIdx.x, lane = tid & 31;
  const int lmod = lane & 15, lhalf = lane >> 4, wave = tid >> 5;
  const int bk = blockIdx.x;  // b*8 + kvh
  const int b = bk >> 3, kvh = bk & 7;
  const int part = blockIdx.y;
  const int tb = part * 2048;

  // Stage Q (async): 64 rows x 128
#pragma unroll
  for (int i = 0; i < 8; ++i) {
    int idx = tid + 256 * i;
    int row = idx >> 5, c4 = idx & 31;
    int rep = row >> 4, s = row & 15;
    async_ld_b128(lds_off(&q_lds[row * QS + c4 * 4]),
                  q + ((((size_t)b * 16 + s) * 32) + kvh * 4 + rep) * 128 + c4 * 4);
  }
  if (tid < 64) { m_l[tid] = -3.0e38f; l_l[tid] = 0.0f; }

  v8f o[4] = {v8f{}, v8f{}, v8f{}, v8f{}};
  const float scale = 0.08838834764831845f;  // 1/sqrt(128)

  for (int t0 = tb; t0 < tb + 2048; t0 += 64) {
    __syncthreads();
    // ---- async stage: all K first, then all V (completion is in-order, so
    // waiting down to 8 outstanding leaves only V in flight)
#pragma unroll
    for (int i = 0; i < 8; ++i) {
      int idx = tid + 256 * i;
      int row = idx >> 5, c4 = idx & 31;
      int t = t0 + row;
      const float* ksrc =
          (t < 4080)
              ? cache_k + (((size_t)b * 4096 + t) * 8 + kvh) * 128 + c4 * 4
              : kws + (((size_t)b * 16 + (t - 4080)) * 8 + kvh) * 128 + c4 * 4;
      async_ld_b128_nt(lds_off(&k_lds[row * QS + c4 * 4]), ksrc);
    }
#pragma unroll
    for (int i = 0; i < 8; ++i) {
      int idx = tid + 256 * i;
      int row = idx >> 5, c4 = idx & 31;
      int t = t0 + row;
      const float* vsrc =
          (t < 4080)
              ? cache_v + (((size_t)b * 4096 + t) * 8 + kvh) * 128 + c4 * 4
              : vws + (((size_t)b * 16 + (t - 4080)) * 8 + kvh) * 128 + c4 * 4;
      async_ld_b128_nt(lds_off(&v_lds[row * VS + c4 * 4]), vsrc);
    }
    wait_async8();  // Q + K resident; V still streaming during stage 1/2
    __syncthreads();

    // ---- stage 1: S = Q K^T (64x64)
    {
      const int mw = wave >> 1;
      const int nb = (wave & 1) * 2;
      v8f s0 = {}, s1 = {};
#pragma unroll
      for (int k4 = 0; k4 < 128; k4 += 4) {
        const int kk = k4 + 2 * lhalf;
        v2f a = *(const v2f*)&q_lds[(mw * 16 + lmod) * QS + kk];
        v2f b0 = *(const v2f*)&k_lds[(nb * 16 + lmod) * QS + kk];
        v2f b1 = *(const v2f*)&k_lds[((nb + 1) * 16 + lmod) * QS + kk];
        s0 = wmma4(a, b0, s0);
        s1 = wmma4(a, b1, s1);
      }
#pragma unroll
      for (int r = 0; r < 8; ++r) {
        int row = mw * 16 + r + lhalf * 8;
        s_lds[row * SS + nb * 16 + lmod] = s0[r];
        s_lds[row * SS + (nb + 1) * 16 + lmod] = s1[r];
      }
    }
    __syncthreads();

    // ---- stage 2: online softmax, 4 threads per row (interleaved columns,
    // conflict-free banks), shfl-xor reductions; all 256 threads active.
    {
      const int r = tid >> 2, sub = tid & 3, s = r & 15;
      float vloc[16];
      float cmax = -3.0e38f;
#pragma unroll
      for (int jj = 0; jj < 16; ++jj) {
        int j = sub + 4 * jj;
        int t = t0 + j;
        float val = s_lds[r * SS + j];
        val = (t <= 4080 + s) ? val * scale : (val - 1.0e9f) * scale;
        vloc[jj] = val;
        cmax = fmaxf(cmax, val);
      }
      cmax = fmaxf(cmax, __shfl_xor(cmax, 1, 32));
      cmax = fmaxf(cmax, __shfl_xor(cmax, 2, 32));
      float mold = m_l[r];  // lockstep: read happens before sub0's update
      float newm = fmaxf(mold, cmax);
      float sum = 0.0f;
#pragma unroll
      for (int jj = 0; jj < 16; ++jj) {
        float pv = __expf(vloc[jj] - newm);
        s_lds[r * SS + sub + 4 * jj] = pv;
        sum += pv;
      }
      sum += __shfl_xor(sum, 1, 32);
      sum += __shfl_xor(sum, 2, 32);
      if (sub == 0) {
        float corr = __expf(mold - newm);
        m_l[r] = newm;
        l_l[r] = l_l[r] * corr + sum;
        c_l[r] = corr;
      }
    }
    wait_async0();  // V resident for everyone after the next barrier
    __syncthreads();

    // ---- stage 3: O = O*corr + P @ V ; wave owns HD cols [16w,16w+16)
#pragma unroll
    for (int mt = 0; mt < 4; ++mt)
#pragma unroll
      for (int r = 0; r < 8; ++r) {
        int row = mt * 16 + r + lhalf * 8;
        o[mt][r] *= c_l[row];
      }
#pragma unroll
    for (int k4 = 0; k4 < 64; k4 += 4) {
      const int kk = k4 + 2 * lhalf;
      v2f bb;
      bb.x = v_lds[kk * VS + wave * 16 + lmod];
      bb.y = v_lds[(kk + 1) * VS + wave * 16 + lmod];
#pragma unroll
      for (int mt = 0; mt < 4; ++mt) {
        v2f a = *(const v2f*)&s_lds[(mt * 16 + lmod) * SS + kk];
        o[mt] = wmma4(a, bb, o[mt]);
      }
    }
  }

  // ---- emit partials (unnormalized O; per-row m, l)
  const size_t obase = ((size_t)(part * 128 + bk) * 64) * 128;
#pragma unroll
  for (int mt = 0; mt < 4; ++mt)
#pragma unroll
    for (int r = 0; r < 8; ++r) {
      int row = mt * 16 + r + lhalf * 8;
      o_part[obase + (size_t)row * 128 + wave * 16 + lmod] = o[mt][r];
    }
  if (tid < 64) {
    m_part[(part * 128 + bk) * 64 + tid] = m_l[tid];
    l_part[(part * 128 + bk) * 64 + tid] = l_l[tid];
  }
}

// Deterministic combine of the two key-split partials -> [B,S,H,HD]
__global__ __launch_bounds__(128)
void attn_combine(const float* __restrict__ o_part,
                  const float* __restrict__ m_part,
                  const float* __restrict__ l_part, float* __restrict__ out) {
  int bk = blockIdx.x >> 6;
  int row = blockIdx.x & 63;
  int col = threadIdx.x;
  int b = bk >> 3, kvh = bk & 7;
  int rep = row >> 4, s = row & 15;
  float m0 = m_part[bk * 64 + row], m1 = m_part[8192 + bk * 64 + row];
  float l0 = l_part[bk * 64 + row], l1 = l_part[8192 + bk * 64 + row];
  float m = fmaxf(m0, m1);
  float s0 = __expf(m0 - m), s1 = __expf(m1 - m);
  float o0 = o_part[((size_t)bk * 64 + row) * 128 + col];
  float o1 = o_part[((size_t)(128 + bk) * 64 + row) * 128 + col];
  float val = (o0 * s0 + o1 * s1) / (l0 * s0 + l1 * s1);
  out[((((size_t)b * 16 + s) * 32) + kvh * 4 + rep) * 128 + col] = val;
}

// ---------------------------------------------------------------------------
// Host launcher
// ---------------------------------------------------------------------------
extern "C" void kernel_launch(void* const* d_in, const int* in_sizes, int n_in,
                              void* d_out, int out_size, void* d_ws,
                              size_t ws_size, hipStream_t stream) {
  const float* x       = (const float*)d_in[0];
  const float* fc      = (const float*)d_in[1];
  const float* fs      = (const float*)d_in[2];
  const float* cache_k = (const float*)d_in[4];
  const float* cache_v = (const float*)d_in[5];
  const float* wq      = (const float*)d_in[6];
  const float* bq      = (const float*)d_in[7];
  const float* wk      = (const float*)d_in[8];
  const float* bk      = (const float*)d_in[9];
  const float* wv      = (const float*)d_in[10];
  const float* bv      = (const float*)d_in[11];
  const float* wo      = (const float*)d_in[12];
  const float* bo      = (const float*)d_in[13];
  float* out = (float*)d_out;

  float* ws     = (float*)d_ws;
  float* p_ws   = ws;                      // 2*256*4096 partials (8 MB)
  float* q_ws   = p_ws + 2 * 256 * 4096;   // 256*4096
  float* k_ws   = q_ws + 256 * 4096;       // 256*1024
  float* v_ws   = k_ws + 256 * 1024;       // 256*1024
  float* a_ws   = v_ws + 256 * 1024;       // 256*4096
  float* o_part = a_ws + 256 * 4096;       // 2*128*64*128 (8 MB)
  float* m_part = o_part + 2 * 128 * 64 * 128;
  float* l_part = m_part + 2 * 128 * 64;

  const size_t attn_lds =
      (size_t)(2 * 64 * 132 + 64 * 144 + 64 * 68 + 3 * 64) * sizeof(float);

  gemm_wmma<<<dim3(64, 2), 256, 0, stream>>>(x, wq, p_ws, 4096);
  reduce_rope<<<(256 * 32 * 64 + 255) / 256, 256, 0, stream>>>(p_ws, bq, q_ws, fc, fs, 32);
  gemm_wmma<<<dim3(16, 2), 256, 0, stream>>>(x, wk, p_ws, 1024);
  reduce_rope<<<(256 * 8 * 64 + 255) / 256, 256, 0, stream>>>(p_ws, bk, k_ws, fc, fs, 8);
  gemm_wmma<<<dim3(16, 2), 256, 0, stream>>>(x, wv, p_ws, 1024);
  reduce_bias<<<(256 * 1024 + 255) / 256, 256, 0, stream>>>(p_ws, bv, v_ws, 1024, 256 * 1024);

  attn_kernel<<<dim3(128, 2), 256, attn_lds, stream>>>(
      q_ws, k_ws, v_ws, cache_k, cache_v, o_part, m_part, l_part);
  attn_combine<<<128 * 64, 128, 0, stream>>>(o_part, m_part, l_part, a_ws);

  gemm_wmma<<<dim3(64, 2), 256, 0, stream>>>(a_ws, wo, p_ws, 4096);
  reduce_bias<<<(256 * 4096 + 255) / 256, 256, 0, stream>>>(p_ws, bo, out, 4096, 256 * 4096);
}